// DFFN_69561290326735
// MI455X (gfx1250) — compile-verified
//
#include <hip/hip_runtime.h>
#include <math.h>

// ---- problem constants (match reference) ----
#define BB   4
#define DIMC 64
#define C2   256
#define HF   128
#define HH   256
#define WW   256
#define SP   (HH*WW)   // 65536 pixels per image plane
#define PP   8

typedef __attribute__((ext_vector_type(2))) float f32x2;
typedef __attribute__((ext_vector_type(8))) float f32x8;

// -----------------------------------------------------------------------------
// Kernel 0: per-channel circular-conv kernel from the rfft2 filter.
// irfft2(rfft2(p) * filt) == circular_conv(p, g) with g = ifft2(G),
// G[k][l] = filt[k][l] (l<=4), G[k][l] = filt[(8-k)%8][8-l] (l>4).
// g[a][b] = (1/64) * sum_{k,l} G[k][l] * cos(2*pi*(k*a+l*b)/8)  -- exact table.
// -----------------------------------------------------------------------------
__global__ __launch_bounds__(64) void k_ck(const float* __restrict__ filt,
                                           float* __restrict__ ck) {
  const int c = blockIdx.x;        // 0..255
  const int t = threadIdx.x;       // 0..63
  const int a  = t >> 3;
  const int b8 = t & 7;
  const float R = 0.70710678118654752f;
  const float ctab[8] = {1.f, R, 0.f, -R, -1.f, -R, 0.f, R}; // cos(2*pi*m/8)
  const float* fc = filt + c * 40; // (8 x 5) per channel
  float s = 0.f;
  for (int k = 0; k < 8; ++k)
    for (int l = 0; l < 8; ++l) {
      float Gv = (l <= 4) ? fc[k * 5 + l] : fc[((8 - k) & 7) * 5 + (8 - l)];
      s += Gv * ctab[(k * a + l * b8) & 7];
    }
  ck[c * 64 + t] = s * (1.0f / 64.0f);
}

// -----------------------------------------------------------------------------
// Kernel 1: project_in GEMM  h[b,o,s] = sum_c w_in[o,c] * x[b,c,s]
// fp32 WMMA, 64x32 register block per wave (4 M-tiles x 2 N-tiles, 8 accs).
// ISA f32 layouts: A 16x4 -> lane M=lane&15, K = vgpr + 2*(lane>>4);
//                  B  4x16 -> lane N=lane&15, K = vgpr + 2*(lane>>4);
//                  C/D row  = vgpr + 8*(lane>>4), col = lane&15.
// -----------------------------------------------------------------------------
__global__ __launch_bounds__(256) void k_proj_in(const float* __restrict__ x,
                                                 const float* __restrict__ w_in,
                                                 float* __restrict__ h) {
  const int lane  = threadIdx.x & 31;
  const int wave  = threadIdx.x >> 5;
  const int hi    = lane >> 4;
  const int lo    = lane & 15;
  const int mBase = (blockIdx.y * 2 + (wave & 1)) * 64;     // 4 M-tiles from here
  const int n0    = (blockIdx.x * 4 + (wave >> 1)) * 32;    // 2 N-tiles from here
  const int b     = blockIdx.z;
  const float* xb = x + (size_t)b * DIMC * SP;

  f32x8 acc[4][2];
#pragma unroll
  for (int mi = 0; mi < 4; ++mi)
#pragma unroll
    for (int ni = 0; ni < 2; ++ni)
      acc[mi][ni] = (f32x8){0.f,0.f,0.f,0.f,0.f,0.f,0.f,0.f};

#pragma unroll
  for (int k0 = 0; k0 < DIMC; k0 += 4) {
    const int kk = k0 + 2 * hi;
    f32x2 Bf[2];
#pragma unroll
    for (int ni = 0; ni < 2; ++ni) {
      const int n = n0 + ni * 16 + lo;
      Bf[ni].x = xb[(size_t)(kk + 0) * SP + n];
      Bf[ni].y = xb[(size_t)(kk + 1) * SP + n];
    }
#pragma unroll
    for (int mi = 0; mi < 4; ++mi) {
      f32x2 A = *(const f32x2*)(w_in + (mBase + mi * 16 + lo) * DIMC + kk);
#pragma unroll
      for (int ni = 0; ni < 2; ++ni)
        acc[mi][ni] = __builtin_amdgcn_wmma_f32_16x16x4_f32(
            false, A, false, Bf[ni], (short)0, acc[mi][ni], false, false);
    }
  }

#pragma unroll
  for (int mi = 0; mi < 4; ++mi)
#pragma unroll
    for (int ni = 0; ni < 2; ++ni) {
      float* hb = h + ((size_t)b * C2 + mBase + mi * 16 + 8 * hi) * SP
                    + n0 + ni * 16 + lo;
#pragma unroll
      for (int r = 0; r < 8; ++r)
        hb[(size_t)r * SP] = acc[mi][ni][r];
    }
}

// -----------------------------------------------------------------------------
// Kernel 2: in-place patchwise spectral gating == 8x8 circular conv per channel.
// One wave per 8x8 patch; 8 patches of the same (b,c,patch-row) per block so
// the 64-float circular kernel is shared in LDS. Patch reads are wave-uniform
// LDS broadcasts; gk reads hit distinct banks.
// -----------------------------------------------------------------------------
__global__ __launch_bounds__(256) void k_spectral(float* __restrict__ h,
                                                  const float* __restrict__ ck) {
  __shared__ float gk[64];
  __shared__ float patch[8][64];
  const unsigned bid = blockIdx.x;          // B*C2*32*4 = 131072 blocks
  const int grp = bid & 3;
  const int py  = (bid >> 2) & 31;
  const int c   = (bid >> 7) & 255;
  const int b   = bid >> 15;
  const int wave = threadIdx.x >> 5;
  const int lane = threadIdx.x & 31;

  if (threadIdx.x < 64) gk[threadIdx.x] = ck[c * 64 + threadIdx.x];

  const int px = grp * 8 + wave;            // patch column 0..31
  float* base = h + ((size_t)b * C2 + c) * SP + (size_t)(py * PP) * WW + px * PP;

  const int i0 = lane, i1 = lane + 32;
  patch[wave][i0] = base[(i0 >> 3) * WW + (i0 & 7)];
  patch[wave][i1] = base[(i1 >> 3) * WW + (i1 & 7)];
  __syncthreads();

  const int v0 = i0 >> 3, u0 = i0 & 7, v1 = i1 >> 3, u1 = i1 & 7;
  float acc0 = 0.f, acc1 = 0.f;
#pragma unroll
  for (int y = 0; y < 8; ++y)
#pragma unroll
    for (int xx = 0; xx < 8; ++xx) {
      float iv = patch[wave][y * 8 + xx];
      acc0 += iv * gk[((v0 - y) & 7) * 8 + ((u0 - xx) & 7)];
      acc1 += iv * gk[((v1 - y) & 7) * 8 + ((u1 - xx) & 7)];
    }
  base[v0 * WW + u0] = acc0;
  base[v1 * WW + u1] = acc1;
}

// -----------------------------------------------------------------------------
// Kernel 3: depthwise 3x3 (SAME, cross-correlation like XLA) on both channel
// halves + exact GELU gate:  g = gelu(dw(h[:128])) * dw(h[128:])
// -----------------------------------------------------------------------------
__global__ __launch_bounds__(256) void k_dwgate(const float* __restrict__ h,
                                                const float* __restrict__ w_dw,
                                                float* __restrict__ g) {
  const unsigned n = blockIdx.x * 256u + threadIdx.x;   // B*HF*SP threads
  const int xx = n & 255;
  const int yy = (n >> 8) & 255;
  const int c  = (n >> 16) & 127;
  const int b  = n >> 23;
  const float* h1 = h + ((size_t)b * C2 + c) * SP;
  const float* h2 = h + ((size_t)b * C2 + c + HF) * SP;
  const float* k1 = w_dw + c * 9;
  const float* k2 = w_dw + (c + HF) * 9;
  float a1 = 0.f, a2 = 0.f;
#pragma unroll
  for (int i = 0; i < 3; ++i) {
    const int y = yy + i - 1;
    if ((unsigned)y >= 256u) continue;
#pragma unroll
    for (int j = 0; j < 3; ++j) {
      const int x = xx + j - 1;
      if ((unsigned)x >= 256u) continue;
      a1 += h1[y * 256 + x] * k1[i * 3 + j];
      a2 += h2[y * 256 + x] * k2[i * 3 + j];
    }
  }
  const float ge = 0.5f * a1 * (1.0f + erff(a1 * 0.70710678118654752f));
  g[((size_t)b * HF + c) * SP + yy * 256 + xx] = ge * a2;
}

// -----------------------------------------------------------------------------
// Kernel 4: project_out GEMM  out[b,o,s] = sum_c w_out[o,c] * g[b,c,s]
// 4 M-tiles cover all of M=64; each wave does 64x32 (8 accs), K=128.
// -----------------------------------------------------------------------------
__global__ __launch_bounds__(256) void k_proj_out(const float* __restrict__ g,
                                                  const float* __restrict__ w_out,
                                                  float* __restrict__ out) {
  const int lane  = threadIdx.x & 31;
  const int wave  = threadIdx.x >> 5;
  const int hi    = lane >> 4;
  const int lo    = lane & 15;
  const int n0    = (blockIdx.x * 8 + wave) * 32;           // 2 N-tiles from here
  const int b     = blockIdx.z;
  const float* gb = g + (size_t)b * HF * SP;

  f32x8 acc[4][2];
#pragma unroll
  for (int mi = 0; mi < 4; ++mi)
#pragma unroll
    for (int ni = 0; ni < 2; ++ni)
      acc[mi][ni] = (f32x8){0.f,0.f,0.f,0.f,0.f,0.f,0.f,0.f};

#pragma unroll
  for (int k0 = 0; k0 < HF; k0 += 4) {
    const int kk = k0 + 2 * hi;
    f32x2 Bf[2];
#pragma unroll
    for (int ni = 0; ni < 2; ++ni) {
      const int n = n0 + ni * 16 + lo;
      Bf[ni].x = gb[(size_t)(kk + 0) * SP + n];
      Bf[ni].y = gb[(size_t)(kk + 1) * SP + n];
    }
#pragma unroll
    for (int mi = 0; mi < 4; ++mi) {
      f32x2 A = *(const f32x2*)(w_out + (mi * 16 + lo) * HF + kk);
#pragma unroll
      for (int ni = 0; ni < 2; ++ni)
        acc[mi][ni] = __builtin_amdgcn_wmma_f32_16x16x4_f32(
            false, A, false, Bf[ni], (short)0, acc[mi][ni], false, false);
    }
  }

#pragma unroll
  for (int mi = 0; mi < 4; ++mi)
#pragma unroll
    for (int ni = 0; ni < 2; ++ni) {
      float* ob = out + ((size_t)b * DIMC + mi * 16 + 8 * hi) * SP
                      + n0 + ni * 16 + lo;
#pragma unroll
      for (int r = 0; r < 8; ++r)
        ob[(size_t)r * SP] = acc[mi][ni][r];
    }
}

// -----------------------------------------------------------------------------
extern "C" void kernel_launch(void* const* d_in, const int* in_sizes, int n_in,
                              void* d_out, int out_size, void* d_ws, size_t ws_size,
                              hipStream_t stream) {
  const float* x        = (const float*)d_in[0];  // (4,64,256,256)
  const float* w_in     = (const float*)d_in[1];  // (256,64)
  const float* w_dw     = (const float*)d_in[2];  // (256,1,3,3)
  const float* fft_filt = (const float*)d_in[3];  // (256,1,1,8,5)
  const float* w_out    = (const float*)d_in[4];  // (64,128)
  float* out = (float*)d_out;                     // (4,64,256,256)

  float* h  = (float*)d_ws;                       // (4,256,65536) = 256 MB
  float* g  = h + (size_t)BB * C2 * SP;           // (4,128,65536) = 128 MB
  float* ck = g + (size_t)BB * HF * SP;           // (256,64)       =  64 KB

  k_ck      <<<256, 64, 0, stream>>>(fft_filt, ck);
  k_proj_in <<<dim3(SP / 128, 2, BB), 256, 0, stream>>>(x, w_in, h);   // (512,2,4)
  k_spectral<<<BB * C2 * 32 * 4, 256, 0, stream>>>(h, ck);
  k_dwgate  <<<(BB * HF * SP) / 256, 256, 0, stream>>>(h, w_dw, g);
  k_proj_out<<<dim3(SP / 256, 1, BB), 256, 0, stream>>>(g, w_out, out); // (256,1,4)
}